// FeatureDistillKL_60833916781214
// MI455X (gfx1250) — compile-verified
//
#include <hip/hip_runtime.h>
#include <cstdint>
#include <cstddef>

// ---------------- problem constants ----------------
#define NSAMP        64
#define BINS         256
#define SAMPLE_ELEMS (256 * 32 * 32)          // 262144 elements per sample
#define EPS          1e-8f
#define INV_T        0.25f                    // 1/T, T=4
#define SCALE        0.25f                    // T^2 / N = 16/64

// ---------------- stage-1 geometry ----------------
#define BLOCK             256                 // 8 waves (wave32)
#define CHUNK             16384               // elements per block (64 KB)
#define TILE_ELEMS        1024                // 4 KB LDS tile = BLOCK * float4
#define NTILES            (CHUNK / TILE_ELEMS)          // 16
#define BLOCKS_PER_SAMPLE (SAMPLE_ELEMS / CHUNK)        // 16

// ---------------- feature probes ----------------
#if defined(__has_builtin)
#  if __has_builtin(__builtin_amdgcn_global_load_async_to_lds_b128)
#    define HAVE_ASYNC 1
#  endif
#  if __has_builtin(__builtin_amdgcn_wmma_f32_16x16x4_f32)
#    define HAVE_WMMA4 1
#  endif
#endif

typedef float v2f __attribute__((ext_vector_type(2)));
typedef float v8f __attribute__((ext_vector_type(8)));
typedef int   gi4 __attribute__((vector_size(16)));   // matches builtin param type

#if defined(HAVE_ASYNC)
typedef __attribute__((address_space(1))) gi4 gi4_g;  // global
typedef __attribute__((address_space(3))) gi4 gi4_l;  // LDS

__device__ __forceinline__ void async_cp16(void* lds, const void* gl) {
  // one b128 per lane: global -> LDS, no VGPR staging, tracked by ASYNCcnt
  __builtin_amdgcn_global_load_async_to_lds_b128(
      (gi4_g*)(uintptr_t)gl,
      (gi4_l*)(uint32_t)(uintptr_t)lds,
      0, 0);
}
__device__ __forceinline__ void wait_async_le1() {
#if __has_builtin(__builtin_amdgcn_s_wait_asynccnt)
  __builtin_amdgcn_s_wait_asynccnt(1);
#else
  asm volatile("s_wait_asynccnt 1" ::: "memory");
#endif
}
__device__ __forceinline__ void wait_async_le0() {
#if __has_builtin(__builtin_amdgcn_s_wait_asynccnt)
  __builtin_amdgcn_s_wait_asynccnt(0);
#else
  asm volatile("s_wait_asynccnt 0" ::: "memory");
#endif
}
#endif

// ---------------- stage 0: zero workspace + output ----------------
__global__ void fdkl_zero_kernel(unsigned* __restrict__ ghist, float* __restrict__ out) {
  const int i = blockIdx.x * blockDim.x + threadIdx.x;
  if (i < 2 * NSAMP * BINS) ghist[i] = 0u;
  if (i == 0) out[0] = 0.0f;
}

// ---------------- stage 1: per-sample histograms ----------------
// grid = (BLOCKS_PER_SAMPLE, NSAMP, 2), block = 256.
// LDS-privatized 256-bin histogram; data streamed via async global->LDS DMA
// (double-buffered 4 KB tiles) when available.
__global__ void __launch_bounds__(BLOCK)
fdkl_hist_kernel(const float* __restrict__ fs, const float* __restrict__ ft,
                 unsigned* __restrict__ ghist) {
  __shared__ float4   buf[2][BLOCK];   // 2 x 4 KB staging tiles
  __shared__ unsigned hist[BINS];

  const int tid    = threadIdx.x;
  const int sample = blockIdx.y;
  const int tensor = blockIdx.z;
  const float*  src  = tensor ? ft : fs;
  const float4* gsrc = (const float4*)(src + (size_t)sample * SAMPLE_ELEMS
                                           + (size_t)blockIdx.x * CHUNK);

  hist[tid] = 0u;                      // BLOCK == BINS
  __syncthreads();

#if defined(HAVE_ASYNC)
  async_cp16(&buf[0][tid], gsrc + tid);                       // prefetch tile 0
  for (int t = 0; t < NTILES; ++t) {
    const int cur = t & 1;
    if (t + 1 < NTILES) {
      async_cp16(&buf[cur ^ 1][tid], gsrc + (size_t)(t + 1) * BLOCK + tid);
      wait_async_le1();                // oldest (tile t) complete for this wave
    } else {
      wait_async_le0();
    }
    __syncthreads();                   // all waves' portions of tile t landed
    const float4 v = buf[cur][tid];
    atomicAdd(&hist[(unsigned)(int)v.x & 255u], 1u);
    atomicAdd(&hist[(unsigned)(int)v.y & 255u], 1u);
    atomicAdd(&hist[(unsigned)(int)v.z & 255u], 1u);
    atomicAdd(&hist[(unsigned)(int)v.w & 255u], 1u);
    __syncthreads();                   // done reading buf[cur] before it refills
  }
#else
  for (int t = 0; t < NTILES; ++t) {
    const float4 v = gsrc[(size_t)t * BLOCK + tid];
    atomicAdd(&hist[(unsigned)(int)v.x & 255u], 1u);
    atomicAdd(&hist[(unsigned)(int)v.y & 255u], 1u);
    atomicAdd(&hist[(unsigned)(int)v.z & 255u], 1u);
    atomicAdd(&hist[(unsigned)(int)v.w & 255u], 1u);
  }
#endif

  __syncthreads();
  atomicAdd(&ghist[((size_t)tensor * NSAMP + sample) * BINS + tid], hist[tid]);
}

// ---------------- stage 2: KL over 256 bins per sample ----------------
// One wave (32 threads) per 16 samples, grid = 4.
// z = log(h+eps)/T in [-4.6, 3.2]  =>  exp(z) = (h+eps)^(1/4) is safe in f32,
// so logsumexp is max-free and every reduction is a row-sum = A x ones,
// done with v_wmma_f32_16x16x4_f32 (16x4 A chunk, all-ones B, f32 accum D).
__global__ void __launch_bounds__(32)
fdkl_kl_kernel(const unsigned* __restrict__ ghist, float* __restrict__ out) {
  const int lane  = threadIdx.x & 31;
  const int i0    = blockIdx.x << 4;           // 16 samples per block
  const int row   = i0 + (lane & 15);          // A-matrix: M = lane&15
  const int khalf = (lane >> 4) << 1;          // lanes 0-15: K+{0,1}; 16-31: K+{2,3}
  const unsigned* hs = ghist + (size_t)row * BINS;
  const unsigned* ht = ghist + (size_t)(NSAMP + row) * BINS;

#if defined(HAVE_WMMA4)
  __shared__ float Ss[16];
  __shared__ float St[16];
  v8f accS = {0.f, 0.f, 0.f, 0.f, 0.f, 0.f, 0.f, 0.f};
  v8f accT = {0.f, 0.f, 0.f, 0.f, 0.f, 0.f, 0.f, 0.f};
  v2f ones = {1.0f, 1.0f};

  // pass A: row sums of exp(z_s), exp(z_t) via D += A x ones
  for (int c = 0; c < BINS / 4; ++c) {
    const int k = (c << 2) + khalf;
    const float h0 = (float)hs[k] + EPS, h1 = (float)hs[k + 1] + EPS;
    const float g0 = (float)ht[k] + EPS, g1 = (float)ht[k + 1] + EPS;
    v2f as = {sqrtf(sqrtf(h0)), sqrtf(sqrtf(h1))};   // exp(log(h)/4) = h^0.25
    v2f at = {sqrtf(sqrtf(g0)), sqrtf(sqrtf(g1))};
    accS = __builtin_amdgcn_wmma_f32_16x16x4_f32(false, as, false, ones,
                                                 (short)0, accS, false, false);
    accT = __builtin_amdgcn_wmma_f32_16x16x4_f32(false, at, false, ones,
                                                 (short)0, accT, false, false);
  }
  // D layout: lanes 0-15 VGPR v = row v; lanes 16-31 VGPR v = row v+8
  if (lane == 0)
    for (int v = 0; v < 8; ++v) { Ss[v] = accS[v]; St[v] = accT[v]; }
  if (lane == 16)
    for (int v = 0; v < 8; ++v) { Ss[8 + v] = accS[v]; St[8 + v] = accT[v]; }
  __syncthreads();

  const float S_s = Ss[lane & 15];
  const float S_t = St[lane & 15];
  const float ls = logf(S_s), lt = logf(S_t), invSt = 1.0f / S_t;

  // pass B: KL terms, row-summed by WMMA as well
  v8f accK = {0.f, 0.f, 0.f, 0.f, 0.f, 0.f, 0.f, 0.f};
  for (int c = 0; c < BINS / 4; ++c) {
    const int k = (c << 2) + khalf;
    const float h0 = (float)hs[k] + EPS, h1 = (float)hs[k + 1] + EPS;
    const float g0 = (float)ht[k] + EPS, g1 = (float)ht[k + 1] + EPS;
    const float zs0 = logf(h0) * INV_T, zs1 = logf(h1) * INV_T;
    const float zt0 = logf(g0) * INV_T, zt1 = logf(g1) * INV_T;
    const float et0 = sqrtf(sqrtf(g0)), et1 = sqrtf(sqrtf(g1));
    v2f tv = {et0 * invSt * (zt0 - lt - zs0 + ls),
              et1 * invSt * (zt1 - lt - zs1 + ls)};
    accK = __builtin_amdgcn_wmma_f32_16x16x4_f32(false, tv, false, ones,
                                                 (short)0, accK, false, false);
  }
  float part = 0.f;
  for (int v = 0; v < 8; ++v) part += accK[v];   // rows of this half (any column)
  const float tot = part + __shfl_xor(part, 16); // add other half's 8 rows
  if (lane == 0) atomicAdd(out, tot * SCALE);
#else
  // scalar fallback (identical math, shuffle reductions)
  float sS = 0.f, sT = 0.f;
  for (int c = 0; c < BINS / 4; ++c) {
    const int k = (c << 2) + khalf;
    sS += sqrtf(sqrtf((float)hs[k] + EPS)) + sqrtf(sqrtf((float)hs[k + 1] + EPS));
    sT += sqrtf(sqrtf((float)ht[k] + EPS)) + sqrtf(sqrtf((float)ht[k + 1] + EPS));
  }
  const float S_s = sS + __shfl_xor(sS, 16);
  const float S_t = sT + __shfl_xor(sT, 16);
  const float ls = logf(S_s), lt = logf(S_t), invSt = 1.0f / S_t;
  float kpart = 0.f;
  for (int c = 0; c < BINS / 4; ++c) {
    const int k = (c << 2) + khalf;
    const float h0 = (float)hs[k] + EPS, h1 = (float)hs[k + 1] + EPS;
    const float g0 = (float)ht[k] + EPS, g1 = (float)ht[k + 1] + EPS;
    const float zs0 = logf(h0) * INV_T, zs1 = logf(h1) * INV_T;
    const float zt0 = logf(g0) * INV_T, zt1 = logf(g1) * INV_T;
    kpart += sqrtf(sqrtf(g0)) * invSt * (zt0 - lt - zs0 + ls);
    kpart += sqrtf(sqrtf(g1)) * invSt * (zt1 - lt - zs1 + ls);
  }
  for (int off = 16; off; off >>= 1) kpart += __shfl_xor(kpart, off);
  if (lane == 0) atomicAdd(out, kpart * SCALE);
#endif
}

// ---------------- launcher ----------------
extern "C" void kernel_launch(void* const* d_in, const int* in_sizes, int n_in,
                              void* d_out, int out_size, void* d_ws, size_t ws_size,
                              hipStream_t stream) {
  const float* fs = (const float*)d_in[0];
  const float* ft = (const float*)d_in[1];
  float* out = (float*)d_out;
  unsigned* ghist = (unsigned*)d_ws;   // 2 * 64 * 256 u32 = 128 KB

  fdkl_zero_kernel<<<(2 * NSAMP * BINS + 255) / 256, 256, 0, stream>>>(ghist, out);

  dim3 g1(BLOCKS_PER_SAMPLE, NSAMP, 2);
  fdkl_hist_kernel<<<g1, BLOCK, 0, stream>>>(fs, ft, ghist);

  fdkl_kl_kernel<<<NSAMP / 16, 32, 0, stream>>>(ghist, out);
}